// StructuralSparseBlockT21_76673756168869
// MI455X (gfx1250) — compile-verified
//
#include <hip/hip_runtime.h>

typedef __attribute__((ext_vector_type(2))) float v2f;
typedef __attribute__((ext_vector_type(8))) float v8f;
typedef unsigned int v4u __attribute__((ext_vector_type(4)));
typedef int v4i __attribute__((ext_vector_type(4)));
typedef int v8i __attribute__((ext_vector_type(8)));

constexpr int kE  = 512;   // embedding
constexpr int kT  = 32;    // tokens
constexpr int kH  = 256;   // hidden
constexpr int kIT = 5;     // inhibition iterations
constexpr int kB  = 4096;  // batch
constexpr int kKC = 16;    // K-chunk staged in LDS
constexpr int kKP = 544;   // floats per kpair row (512 + 32 pad -> disjoint bank halves)
#define EPS 1e-6f

// Each block: one token t, 16 batch rows. 8 waves; wave w owns a 16x32 output
// strip (cols w*32..w*32+31) of BOTH the hidden and the value GEMM.
// fp32 WMMA: V_WMMA_F32_16X16X4_F32, accumulate over E=512 in steps of 4.
// Weights staged in LDS K-pair interleaved: lds[kpair*kKP + n*2 + (k&1)],
// so every B fragment is one 8-byte aligned ds_load_b64 (no VGPR repacking).
// The x tile (16x16 fp32, row stride E) is DMA'd by the Tensor Data Mover.
__global__ __launch_bounds__(256) void ssb_fused_kernel(
    const float* __restrict__ x,       // [B,E]
    const float* __restrict__ w_in,    // [T,E,H]
    const float* __restrict__ w_val,   // [T,E,H]
    const float* __restrict__ alphas,  // [T,IT]
    const float* __restrict__ scales,  // [T,H]
    const float* __restrict__ gamma,   // [T,H]
    const float* __restrict__ beta,    // [T,H]
    float* __restrict__ out)           // [B,T,H]
{
  // [0 .. 8*kKP)        w_in chunk, kpair-interleaved   (reused: hidden tile)
  // [8*kKP .. 16*kKP)   w_val chunk, kpair-interleaved  (reused: value tile)
  // [16*kKP .. +256)    x tile [16][kKC]  (TDM destination)
  __shared__ __align__(16) float smem[16 * kKP + 16 * kKC];
  float* lwi = smem;
  float* lwv = smem + 8 * kKP;
  float* lx  = smem + 16 * kKP;

  const int tid  = threadIdx.x;
  const int lane = tid & 31;
  const int wave = tid >> 5;
  const int t    = blockIdx.y;
  const int row0 = blockIdx.x * 16;
  const int nb   = wave * 32;            // wave's 32-column strip

  const float* wiBase = w_in  + (size_t)t * kE * kH;
  const float* wvBase = w_val + (size_t)t * kE * kH;

  v8f ah0 = {}, ah1 = {}, av0 = {}, av1 = {};

  // A fragment (16x4 f32): lanes 0-15 -> K pair {0,1}, lanes 16-31 -> {2,3}
  const int arow = lane & 15;
  const int akk  = (lane >> 4) * 2;
  // B fragment (4x16 f32): lane-static float offset into interleaved layout
  const int bofs = (lane >> 4) * kKP + (nb + (lane & 15)) * 2;

  // ---- TDM descriptor (uniform) for the x tile: 16x16 f32, row stride E ----
  // Group1 (ISA 8.4): data_size=4B; tensor_dim0=E; tensor_dim1=B;
  //                   tile_dim0=16; tile_dim1=16; tensor_dim0_stride=E.
  const v8i tdmG1 = {
      (int)(2u << 16),                                  // data_size=2 (4B)
      (int)(((unsigned)kE & 0xffffu) << 16),            // tensor_dim0[15:0]
      (int)((((unsigned)kE >> 16) & 0xffffu) |
            (((unsigned)kB & 0xffffu) << 16)),          // dim0[31:16]|dim1[15:0]
      (int)((((unsigned)kB >> 16) & 0xffffu) |
            (16u << 16)),                               // dim1[31:16]|tile_dim0
      (int)(16u),                                       // tile_dim1 | tile_dim2=0
      (int)kE,                                          // tensor_dim0_stride[31:0]
      0, 0 };
  const v4i tdmG2 = {0, 0, 0, 0};                       // 2D tile: dims 2..4 unused
  const v4i tdmG3 = {0, 0, 0, 0};
  const unsigned ldsX = (unsigned)(uintptr_t)(void*)lx; // LDS byte address of x tile
  const unsigned long long xRowBase =
      (unsigned long long)(uintptr_t)x + (unsigned long long)row0 * kE * 4ull;

  for (int kc = 0; kc < kE; kc += kKC) {
    // ---- TDM: DMA x tile [16 rows][kKC floats] into lx (wave 0 issues) ----
    if (wave == 0) {
      unsigned long long ga = xRowBase + (unsigned)kc * 4u;   // tile start
      v4u g0;
      g0.x = 1u;                                   // count=1, user mode
      g0.y = ldsX;                                 // lds_addr
      g0.z = (unsigned)(ga & 0xffffffffull);       // global_addr[31:0]
      g0.w = (unsigned)((ga >> 32) & 0x01ffffffull) | 0x80000000u; // [56:32]|type=2
#if __has_include(<hip/amd_detail/amd_gfx1250_TDM.h>)
      __builtin_amdgcn_tensor_load_to_lds(g0, tdmG1, tdmG2, tdmG3,
                                          (v8i){0, 0, 0, 0, 0, 0, 0, 0}, 0);
#else
      __builtin_amdgcn_tensor_load_to_lds(g0, tdmG1, tdmG2, tdmG3, 0);
#endif
      __builtin_amdgcn_s_wait_tensorcnt(0);
    }
    // stage weight chunks: coalesced float4 global loads, scatter into
    // kpair-interleaved LDS:  dst = kpair*kKP + n*2 + (k&1)
    {
      const float4* s0 = (const float4*)(wiBase + (size_t)kc * kH);
      const float4* s1 = (const float4*)(wvBase + (size_t)kc * kH);
#pragma unroll
      for (int i = 0; i < 4; ++i) {
        int p  = tid + i * 256;          // float4 index in [0,1024)
        int k  = p >> 6;                 // 64 float4 per K-row
        int n  = (p & 63) * 4;
        int d  = (k >> 1) * kKP + n * 2 + (k & 1);
        float4 a4 = s0[p];
        lwi[d + 0] = a4.x; lwi[d + 2] = a4.y; lwi[d + 4] = a4.z; lwi[d + 6] = a4.w;
        float4 b4 = s1[p];
        lwv[d + 0] = b4.x; lwv[d + 2] = b4.y; lwv[d + 4] = b4.z; lwv[d + 6] = b4.w;
      }
    }
    // prefetch next chunk's weights (global_prefetch_b8): 256 threads x 64B
    if (kc + kKC < kE) {
      const char* pw0 = (const char*)(wiBase + (size_t)(kc + kKC) * kH) + tid * 64;
      const char* pw1 = (const char*)(wvBase + (size_t)(kc + kKC) * kH) + tid * 64;
      __builtin_prefetch(pw0, 0, 1);
      __builtin_prefetch(pw1, 0, 1);
    }
    __syncthreads();

#pragma unroll
    for (int kk = 0; kk < kKC; kk += 4) {
      v2f a = *(const v2f*)&lx[arow * kKC + kk + akk];
      int bi = (kk >> 1) * kKP + bofs;
      v2f bi0 = *(const v2f*)&lwi[bi];
      v2f bi1 = *(const v2f*)&lwi[bi + 32];
      v2f bv0 = *(const v2f*)&lwv[bi];
      v2f bv1 = *(const v2f*)&lwv[bi + 32];
      ah0 = __builtin_amdgcn_wmma_f32_16x16x4_f32(false, a, false, bi0, (short)0, ah0, false, false);
      ah1 = __builtin_amdgcn_wmma_f32_16x16x4_f32(false, a, false, bi1, (short)0, ah1, false, false);
      av0 = __builtin_amdgcn_wmma_f32_16x16x4_f32(false, a, false, bv0, (short)0, av0, false, false);
      av1 = __builtin_amdgcn_wmma_f32_16x16x4_f32(false, a, false, bv1, (short)0, av1, false, false);
    }
    __syncthreads();
  }

  // Dump accumulators to LDS tiles: hidden -> smem[0:16*H), value -> next 16*H.
  // C/D layout: VGPR r, lanes 0-15 => M=r, lanes 16-31 => M=r+8; N = lane&15.
  float* lh = smem;
  float* lv = smem + 16 * kH;
  {
    int cr = (lane >> 4) * 8;
    int cc = lane & 15;
#pragma unroll
    for (int r = 0; r < 8; ++r) {
      lh[(cr + r) * kH + nb + cc]      = ah0[r];
      lh[(cr + r) * kH + nb + 16 + cc] = ah1[r];
      lv[(cr + r) * kH + nb + cc]      = av0[r];
      lv[(cr + r) * kH + nb + 16 + cc] = av1[r];
    }
  }
  __syncthreads();

  // Epilogue: each wave owns one row per pass; lane covers cols lane + 32*j.
  for (int pass = 0; pass < 2; ++pass) {
    int row = wave + pass * 8;
    float h[8];
#pragma unroll
    for (int j = 0; j < 8; ++j) {
      float v0 = lh[row * kH + lane + j * 32];
      h[j] = v0 > 0.f ? v0 : 0.f;                       // relu(hidden)
    }
#pragma unroll
    for (int it = 0; it < kIT; ++it) {                  // competitive inhibition
      float s = 0.f;
#pragma unroll
      for (int j = 0; j < 8; ++j) s += h[j];
#pragma unroll
      for (int off = 16; off > 0; off >>= 1) s += __shfl_xor(s, off, 32);
      float m = s * (1.0f / kH);
      float a = alphas[t * kIT + it];
#pragma unroll
      for (int j = 0; j < 8; ++j) {
        float vv = h[j] - a * m;
        h[j] = vv > 0.f ? vv : 0.f;
      }
    }
    float sv[8];
    float sum = 0.f;
#pragma unroll
    for (int j = 0; j < 8; ++j) {                       // gated value path
      int col = lane + j * 32;
      sv[j] = scales[t * kH + col] * h[j] * lv[row * kH + col];
      sum += sv[j];
    }
#pragma unroll
    for (int off = 16; off > 0; off >>= 1) sum += __shfl_xor(sum, off, 32);
    float mu = sum * (1.0f / kH);
    float var = 0.f;
#pragma unroll
    for (int j = 0; j < 8; ++j) { float d = sv[j] - mu; var += d * d; }
#pragma unroll
    for (int off = 16; off > 0; off >>= 1) var += __shfl_xor(var, off, 32);
    var *= (1.0f / kH);
    float inv = rsqrtf(var + EPS);
    size_t ob = ((size_t)(row0 + row) * kT + t) * kH;
#pragma unroll
    for (int j = 0; j < 8; ++j) {                       // LayerNorm + affine
      int col = lane + j * 32;
      out[ob + col] = (sv[j] - mu) * inv * gamma[t * kH + col] + beta[t * kH + col];
    }
  }
}

extern "C" void kernel_launch(void* const* d_in, const int* in_sizes, int n_in,
                              void* d_out, int out_size, void* d_ws, size_t ws_size,
                              hipStream_t stream) {
  const float* x      = (const float*)d_in[0];
  const float* w_in   = (const float*)d_in[1];
  const float* w_val  = (const float*)d_in[2];
  const float* alphas = (const float*)d_in[3];
  const float* scales = (const float*)d_in[4];
  const float* gamma  = (const float*)d_in[5];
  const float* beta   = (const float*)d_in[6];
  float* out = (float*)d_out;
  (void)in_sizes; (void)n_in; (void)out_size; (void)d_ws; (void)ws_size;

  dim3 grid(kB / 16, kT);   // 256 row-blocks x 32 tokens = 8192 blocks
  ssb_fused_kernel<<<grid, 256, 0, stream>>>(x, w_in, w_val, alphas, scales,
                                             gamma, beta, out);
}